// BatchMinigrid_12824772346586
// MI455X (gfx1250) — compile-verified
//
#include <hip/hip_runtime.h>
#include <math.h>

typedef __attribute__((ext_vector_type(2))) float v2f;
typedef __attribute__((ext_vector_type(8))) float v8f;

#define NPIX 49

__device__ __forceinline__ float fast_tanh(float x) {
#if __has_builtin(__builtin_amdgcn_tanhf)
    return __builtin_amdgcn_tanhf(x);
#elif __has_builtin(__builtin_amdgcn_tanh_f32)
    return __builtin_amdgcn_tanh_f32(x);
#else
    float r;
    // CDNA5 native transcendental; v_nop covers the TRANS32 result hazard
    // since the compiler cannot schedule around inline asm.
    asm volatile("v_tanh_f32 %0, %1\n\tv_nop" : "=v"(r) : "v"(x));
    return r;
#endif
}

// One wave handles 16 environments. Conv expressed as 64x64 (padded 49x49)
// banded matrix times 64x16 state, via V_WMMA_F32_16X16X4_F32.
__global__ __launch_bounds__(256) void fpv_wmma_kernel(
    const int* __restrict__ grids,       // [N,25,25,3] int32
    const int* __restrict__ agent_pos,   // [N,2] int32
    const int* __restrict__ agent_dir,   // [N] int32
    const float* __restrict__ weight,    // [9] f32 (3x3)
    int* __restrict__ out,               // [N,7,7,3] int32
    int N)
{
    __shared__ float wtab[9];
    __shared__ float mst[8][64 * 16];    // per-wave mask state, [pixel][env]

    if (threadIdx.x < 9) wtab[threadIdx.x] = weight[threadIdx.x];
    __syncthreads();

    const int wave = threadIdx.x >> 5;
    const int lane = threadIdx.x & 31;
    const int e    = lane & 15;          // env within wave / matrix column
    const int h    = lane >> 4;          // lane-half
    const int envBase = (blockIdx.x * 8 + wave) * 16;
    if (envBase >= N) return;            // whole-wave exit only (N % 16 == 0)
    const int eg = envBase + e;

    float* ms = mst[wave];

    const int p0 = agent_pos[2 * eg];
    const int p1 = agent_pos[2 * eg + 1];
    const int d  = agent_dir[eg];
    const int offy = (d == 0) ? 0  : (d == 1) ? -3 : (d == 2) ? -6 : -3;
    const int offx = (d == 0) ? -3 : (d == 1) ?  0 : (d == 2) ? -3 : -6;

    // Rotation (kmap=[1,2,3,0] of rot90 axes=(2,1)) as linear transform:
    //   gy = basey + cyi*i + cyj*j ;  gx = basex + cxi*i + cxj*j
    const int cyi = (d == 1) ? -1 : (d == 3) ? 1 : 0;
    const int cyj = (d == 0) ? -1 : (d == 2) ? 1 : 0;
    const int cxi = (d == 0) ?  1 : (d == 2) ? -1 : 0;
    const int cxj = (d == 1) ? -1 : (d == 3) ? 1 : 0;
    const int basey = p0 + offy + ((d <= 1) ? 6 : 0);
    const int basex = p1 + offx + ((d == 1 || d == 2) ? 6 : 0);

    // ---- Gather rotated crop (only bytes we need), pack channels 2b each ----
    unsigned packed[8] = {0, 0, 0, 0, 0, 0, 0, 0};
    unsigned closedbits = 0;
#pragma unroll
    for (int s = 0; s < 32; ++s) {
        const int r = s >> 3, v = s & 7;
        const int p = 16 * r + v + 8 * h;        // pixel slot (0..63)
        int c0 = 0, c1 = 0, c2 = 0;
        if (p < NPIX) {
            const int i = p / 7, j = p - 7 * (p / 7);
            const int gy = basey + cyi * i + cyj * j;
            const int gx = basex + cxi * i + cxj * j;
            if ((unsigned)gy < 25u && (unsigned)gx < 25u) {
                const int* gp = grids + ((eg * 25 + gy) * 25 + gx) * 3;
                c0 = gp[0]; c1 = gp[1]; c2 = gp[2];
            } else {
                c0 = 2; c1 = 2; c2 = 2;          // pad value
            }
            const int closed = (c0 == 2) | (c2 == 1);
            closedbits |= (unsigned)closed << s;
        }
        packed[s >> 2] |=
            (unsigned)((c0 & 3) | ((c1 & 3) << 2) | ((c2 & 3) << 4)) << ((s & 3) * 8);
        ms[p * 16 + e] = 0.0f;                   // zero full 64x16 state
    }
    if (lane < 16) ms[27 * 16 + lane] = 1.0f;    // me: delta at pixel (3,6) -> 27

    // ---- Build banded W tiles in A-matrix layout (K = 2h + component) ----
    constexpr int bandR[24] = {0,0,0,0,0,0, 1,1,1,1,1,1,1,1, 2,2,2,2,2,2,2, 3,3,3};
    constexpr int bandC[24] = {0,1,2,3,4,5, 2,3,4,5,6,7,8,9, 6,7,8,9,10,11,12, 10,11,12};

    v2f A[24];
#pragma unroll
    for (int t = 0; t < 24; ++t) {
        const int r = bandR[t], c = bandC[t];
        const int pr = 16 * r + e;
        const int pi = pr / 7, pj = pr - 7 * (pr / 7);
        float av[2];
#pragma unroll
        for (int v = 0; v < 2; ++v) {
            const int q  = 4 * c + 2 * h + v;
            const int qi = q / 7, qj = q - 7 * (q / 7);
            const int dy = qi - pi, dx = qj - pj;
            const bool ok = (pr < NPIX) && (q < NPIX) &&
                            (dy >= -1) && (dy <= 1) && (dx >= -1) && (dx <= 1);
            const int idx = ok ? ((dy + 1) * 3 + (dx + 1)) : 0;
            av[v] = ok ? wtab[idx] : 0.0f;
        }
        A[t].x = av[0];
        A[t].y = av[1];
    }

    // One conv pass: LDS state -> 24 banded WMMAs -> acc[4] (C/D layout).
    auto conv = [&](v8f* acc) {
        v2f B[13];
#pragma unroll
        for (int c = 0; c < 13; ++c) {
            B[c].x = ms[(4 * c + 2 * h + 0) * 16 + e];
            B[c].y = ms[(4 * c + 2 * h + 1) * 16 + e];
        }
        const v8f z = {};
        acc[0] = z; acc[1] = z; acc[2] = z; acc[3] = z;
#pragma unroll
        for (int t = 0; t < 24; ++t) {
            acc[bandR[t]] = __builtin_amdgcn_wmma_f32_16x16x4_f32(
                false, A[t], false, B[bandC[t]], (short)0, acc[bandR[t]],
                false, false);
        }
    };

    // Nonlinearity + write-back of the state (optionally thresholded).
    auto nonlin = [&](v8f* acc, bool thresh) {
#pragma unroll
        for (int r = 0; r < 4; ++r) {
#pragma unroll
            for (int v = 0; v < 8; ++v) {
                const int s = 8 * r + v;
                const int p = 16 * r + v + 8 * h;
                const bool closed = (closedbits >> s) & 1;
                float val = closed ? -0.01f : fast_tanh(acc[r][v]);
                if (thresh) val = (val > 0.0f) ? 1.0f : 0.0f;
                ms[p * 16 + e] = val;
            }
        }
    };

    // ---- 4 masked propagation steps + thresholded 5th step ----
    v8f acc[4];
    for (int step = 0; step < 4; ++step) {
        conv(acc);
        nonlin(acc, false);
    }
    conv(acc);
    nonlin(acc, true);

    // ---- Final conv, mask = (>0), multiply packed crop, store int32 out ----
    conv(acc);
#pragma unroll
    for (int r = 0; r < 4; ++r) {
#pragma unroll
        for (int v = 0; v < 8; ++v) {
            const int s = 8 * r + v;
            const int p = 16 * r + v + 8 * h;
            if (p < NPIX) {
                const bool m = acc[r][v] > 0.0f;
                const unsigned pk = (packed[s >> 2] >> ((s & 3) * 8)) & 0xFFu;
                int* op = out + (eg * NPIX + p) * 3;
                op[0] = m ? (int)(pk & 3u) : 0;
                op[1] = m ? (int)((pk >> 2) & 3u) : 0;
                op[2] = m ? (int)((pk >> 4) & 3u) : 0;
            }
        }
    }
}

extern "C" void kernel_launch(void* const* d_in, const int* in_sizes, int n_in,
                              void* d_out, int out_size, void* d_ws, size_t ws_size,
                              hipStream_t stream) {
    const int*   grids = (const int*)d_in[0];
    const int*   apos  = (const int*)d_in[1];
    const int*   adir  = (const int*)d_in[2];
    const float* w     = (const float*)d_in[3];
    const int N = in_sizes[2];                 // agent_dir has N elements
    const int blocks = (N + 127) / 128;        // 128 envs per 256-thread block
    fpv_wmma_kernel<<<blocks, 256, 0, stream>>>(grids, apos, adir, w,
                                                (int*)d_out, N);
}